// MoleculeGNN_58093727646210
// MI455X (gfx1250) — compile-verified
//
#include <hip/hip_runtime.h>
#include <hip/hip_bf16.h>
#include <stdint.h>

typedef __attribute__((ext_vector_type(2)))  float  v2f;
typedef __attribute__((ext_vector_type(8)))  float  v8f;
typedef __attribute__((ext_vector_type(4)))  int    v4i;

#define AS1 __attribute__((address_space(1)))
#define AS3 __attribute__((address_space(3)))

#define EPS 1e-5f

// ---------------- CDNA5 async global->LDS helpers ----------------

#if defined(__has_builtin)
#if __has_builtin(__builtin_amdgcn_global_load_async_to_lds_b128)
#define HAVE_ASYNC_LDS 1
#endif
#if __has_builtin(__builtin_amdgcn_s_wait_asynccnt)
#define HAVE_WAIT_ASYNC 1
#endif
#endif

__device__ __forceinline__ void cp16_async(const float* g, float* l) {
#ifdef HAVE_ASYNC_LDS
    // param1: AS(1) int4*  (global source; diag-confirmed type)
    // param2: AS(3) int4*  (LDS destination; low 32 bits of flat ptr = LDS offset)
    __builtin_amdgcn_global_load_async_to_lds_b128(
        (AS1 v4i*)(uintptr_t)g,
        (AS3 v4i*)(unsigned)(uintptr_t)l,
        0, 0);
#else
    *(float4*)l = *(const float4*)g;
#endif
}

__device__ __forceinline__ void wait_async(int n) {   // n must be a literal constant
#ifdef HAVE_WAIT_ASYNC
    if (n == 0)      __builtin_amdgcn_s_wait_asynccnt(0);
    else if (n == 4) __builtin_amdgcn_s_wait_asynccnt(4);
    else             __builtin_amdgcn_s_wait_asynccnt(5);
#else
    if (n == 0)      asm volatile("s_wait_asynccnt 0x0" ::: "memory");
    else if (n == 4) asm volatile("s_wait_asynccnt 0x4" ::: "memory");
    else             asm volatile("s_wait_asynccnt 0x5" ::: "memory");
#endif
}

// ---------------- utility kernels ----------------

__global__ void fill_kernel(float* __restrict__ p, float v, long long n) {
    long long t = (long long)blockIdx.x * blockDim.x + threadIdx.x;
    if (t < n) p[t] = v;
}

__global__ void deg_accum_kernel(const int* __restrict__ dst, float* __restrict__ deg, int E) {
    int t = blockIdx.x * blockDim.x + threadIdx.x;
    if (t < E) atomicAdd(&deg[dst[t]], 1.0f);
}

__global__ void rsqrt_kernel(float* __restrict__ p, int n) {
    int t = blockIdx.x * blockDim.x + threadIdx.x;
    if (t < n) p[t] = rsqrtf(p[t]);
}

// ---------------- WMMA GEMM:  C[M,Nc] = act(A[M,K] @ W[K,Nc] + bias) ----------------
// Block = 256 threads (8 waves). Block tile: 16 (M) x 128 (N). K staged in 32-wide
// chunks through LDS with async global->LDS copies, double buffered.
// Requirements: K % 32 == 0, Nc % 128 == 0 (true for all call sites here).

#define KC 32
#define SA_STRIDE 36   // padded A row stride (floats): bank-conflict-free, rows 16B aligned

__global__ __launch_bounds__(256) void gemm_wmma_kernel(
    const float* __restrict__ A, const float* __restrict__ W,
    const float* __restrict__ bias, float* __restrict__ C,
    int M, int K, int Nc, int relu)
{
    __shared__ __align__(16) float sA[2][16 * SA_STRIDE];
    __shared__ __align__(16) float sB[2][KC * 128];

    const int tid  = threadIdx.x;
    const int wave = tid >> 5;
    const int lane = tid & 31;
    const int hh   = lane >> 4;     // lane half
    const int l16  = lane & 15;

    const int tileM   = blockIdx.x * 16;
    const int colBase = blockIdx.y * 128;     // 128-column panel of this block
    const int nb      = wave * 16 + l16;      // column within panel (wave-private 16-wide tile)

    // ---- stage-issue lambda: copy chunk k0 into buffer bi ----
    // A: 16 rows x 32 k  -> threads 0..127 copy one float4 each   (waves 0-3: 1 async op)
    // B: 32 k  x 128 n   -> all 256 threads copy four float4 each (every wave: 4 async ops)
    auto load_stage = [&](int k0, int bi) {
        if (tid < 128) {
            int row = tid >> 3;
            int kq  = (tid & 7) * 4;
            int r   = tileM + row; if (r >= M) r = M - 1;     // clamp (stores guarded later)
            cp16_async(A + (size_t)r * K + (k0 + kq), &sA[bi][row * SA_STRIDE + kq]);
        }
        #pragma unroll
        for (int i = 0; i < 4; ++i) {
            int idx = tid + i * 256;
            int k   = idx >> 5;
            int nq  = (idx & 31) * 4;
            cp16_async(W + (size_t)(k0 + k) * Nc + (colBase + nq), &sB[bi][k * 128 + nq]);
        }
    };

    v8f acc = {};
    const int nchunks = K / KC;

    load_stage(0, 0);

    for (int c = 0; c < nchunks; ++c) {
        const int cur = c & 1;
        if (c + 1 < nchunks) {
            load_stage(c * KC + KC, cur ^ 1);
            // retire previous stage only (async loads complete in order):
            if (wave < 4) wait_async(5); else wait_async(4);
        } else {
            wait_async(0);
        }
        __syncthreads();

        // ---- 8x V_WMMA_F32_16X16X4_F32 on this 32-deep chunk ----
        // A 16x4 frag: VGPR0 = K (lanes0-15) / K+2 (lanes16-31); VGPR1 = K+1 / K+3. B symmetric.
        #pragma unroll
        for (int kk = 0; kk < KC; kk += 4) {
            const int ka = kk + 2 * hh;
            v2f a; a.x = sA[cur][l16 * SA_STRIDE + ka];  a.y = sA[cur][l16 * SA_STRIDE + ka + 1];
            v2f b; b.x = sB[cur][ka * 128 + nb];         b.y = sB[cur][(ka + 1) * 128 + nb];
            acc = __builtin_amdgcn_wmma_f32_16x16x4_f32(
                false, a, false, b, (short)0, acc, false, false);
        }
        __syncthreads();   // all waves done with `cur` before it is refilled
    }

    // C/D layout: VGPR r -> row tileM + r + 8*hh, column = colBase + nb
    const int col = colBase + nb;
    const float bcol = bias ? bias[col] : 0.0f;
    #pragma unroll
    for (int r = 0; r < 8; ++r) {
        int row = tileM + r + 8 * hh;
        if (row < M) {
            float v = acc[r] + bcol;
            if (relu) v = fmaxf(v, 0.0f);
            C[(size_t)row * Nc + col] = v;
        }
    }
}

// ---------------- edge scatter: agg[dst] += hW[src] * isd[src]*isd[dst] ----------------

__global__ void edge_kernel(const int* __restrict__ src, const int* __restrict__ dst,
                            const float* __restrict__ isd, const float* __restrict__ hW,
                            float* __restrict__ agg, int E, int H)
{
    long long tid = (long long)blockIdx.x * blockDim.x + threadIdx.x;
    const int qn = H >> 2;                 // float4 chunks per row
    long long total = (long long)E * qn;
    if (tid >= total) return;
    int e = (int)(tid / qn);
    int q = (int)(tid % qn) * 4;
    int s = src[e], d = dst[e];
    float norm = isd[s] * isd[d];
    const float4 v = *(const float4*)(hW + (size_t)s * H + q);
    float* ap = agg + (size_t)d * H + q;
    atomicAdd(ap + 0, v.x * norm);
    atomicAdd(ap + 1, v.y * norm);
    atomicAdd(ap + 2, v.z * norm);
    atomicAdd(ap + 3, v.w * norm);
}

// ---------------- fused self-loop + bias + BN(eval) + ReLU + residual ----------------

__global__ void post_kernel(float* __restrict__ h, const float* __restrict__ hW,
                            const float* __restrict__ agg, const float* __restrict__ isd,
                            const float* __restrict__ bias,
                            const float* __restrict__ gamma, const float* __restrict__ beta,
                            const float* __restrict__ mean,  const float* __restrict__ var,
                            int N, int H)
{
    long long tid = (long long)blockIdx.x * blockDim.x + threadIdx.x;
    long long total = (long long)N * H;
    if (tid >= total) return;
    int n = (int)(tid / H);
    int f = (int)(tid % H);
    float is = isd[n];
    float v = agg[tid] + hW[tid] * is * is + bias[f];
    v = (v - mean[f]) * rsqrtf(var[f] + EPS) * gamma[f] + beta[f];
    h[tid] = fmaxf(v, 0.0f) + h[tid];
}

// ---------------- graph pooling (mean + max) ----------------

__global__ void cnt_kernel(const int* __restrict__ batch, float* __restrict__ cnt, int N) {
    int t = blockIdx.x * blockDim.x + threadIdx.x;
    if (t < N) atomicAdd(&cnt[batch[t]], 1.0f);
}

__global__ void pool_kernel(const int* __restrict__ batch, const float* __restrict__ h,
                            float* __restrict__ g, int N, int H)
{
    long long tid = (long long)blockIdx.x * blockDim.x + threadIdx.x;
    const int qn = H >> 2;
    long long total = (long long)N * qn;
    if (tid >= total) return;
    int n = (int)(tid / qn);
    int q = (int)(tid % qn) * 4;
    int b = batch[n];
    const float4 v = *(const float4*)(h + (size_t)n * H + q);
    float*    gm = g + (size_t)b * 2 * H + q;                   // mean-sum half
    unsigned* gx = (unsigned*)(g + (size_t)b * 2 * H + H + q);  // max half (h >= 0)
    atomicAdd(gm + 0, v.x); atomicAdd(gm + 1, v.y);
    atomicAdd(gm + 2, v.z); atomicAdd(gm + 3, v.w);
    atomicMax(gx + 0, __float_as_uint(v.x));
    atomicMax(gx + 1, __float_as_uint(v.y));
    atomicMax(gx + 2, __float_as_uint(v.z));
    atomicMax(gx + 3, __float_as_uint(v.w));
}

__global__ void poolfin_kernel(float* __restrict__ g, const float* __restrict__ cnt,
                               int G, int H)
{
    int t = blockIdx.x * blockDim.x + threadIdx.x;
    if (t >= G * H) return;
    int b = t / H, f = t % H;
    g[(size_t)b * 2 * H + f] /= fmaxf(cnt[b], 1.0f);
}

// ---------------- final tiny head: out[G,T] = g2 @ W3 + b3 ----------------

__global__ void head3_kernel(const float* __restrict__ g2, const float* __restrict__ W3,
                             const float* __restrict__ b3, float* __restrict__ out,
                             int G, int Hq, int T)
{
    int t = blockIdx.x * blockDim.x + threadIdx.x;
    if (t >= G * T) return;
    int gi = t / T, tt = t % T;
    float s = b3[tt];
    const float* row = g2 + (size_t)gi * Hq;
    for (int k = 0; k < Hq; ++k) s += row[k] * W3[k * T + tt];
    out[t] = s;
}

// ---------------- launch ----------------

static inline int cdiv(long long a, int b) { return (int)((a + b - 1) / b); }

extern "C" void kernel_launch(void* const* d_in, const int* in_sizes, int n_in,
                              void* d_out, int out_size, void* d_ws, size_t ws_size,
                              hipStream_t stream) {
    const float* x      = (const float*)d_in[0];
    const int*   ei     = (const int*)  d_in[1];
    const int*   batch  = (const int*)  d_in[2];
    const float* projW  = (const float*)d_in[3];
    const float* projb  = (const float*)d_in[4];
    const float* convW  = (const float*)d_in[5];
    const float* convb  = (const float*)d_in[6];
    const float* bng    = (const float*)d_in[7];
    const float* bnb    = (const float*)d_in[8];
    const float* bnm    = (const float*)d_in[9];
    const float* bnv    = (const float*)d_in[10];
    const float* W1     = (const float*)d_in[11];
    const float* b1     = (const float*)d_in[12];
    const float* W2     = (const float*)d_in[13];
    const float* b2     = (const float*)d_in[14];
    const float* W3     = (const float*)d_in[15];
    const float* b3     = (const float*)d_in[16];

    const int N   = in_sizes[2];
    const int E   = in_sizes[1] / 2;
    const int H   = in_sizes[4];          // 256
    const int FIN = in_sizes[0] / N;      // 64
    const int L   = in_sizes[6] / H;      // 4
    const int T   = in_sizes[16];         // 4
    const int G   = out_size / T;         // 2048
    const int H2  = in_sizes[14];         // 128

    // workspace carve-up (floats)
    float* ws = (float*)d_ws;
    size_t off = 0;
    float* h   = ws + off; off += (size_t)N * H;
    float* hW  = ws + off; off += (size_t)N * H;
    float* agg = ws + off; off += (size_t)N * H;
    float* isd = ws + off; off += (size_t)N;
    float* g   = ws + off; off += (size_t)G * 2 * H;
    float* cnt = ws + off; off += (size_t)G;
    float* g1  = ws + off; off += (size_t)G * H;
    float* g2  = ws + off; off += (size_t)G * H2;
    (void)ws_size;

    const int B = 256;
    const int* src = ei;
    const int* dst = ei + E;

    // 1) degrees with self-loops -> inv_sqrt_deg (in place in isd)
    fill_kernel<<<cdiv(N, B), B, 0, stream>>>(isd, 1.0f, N);
    deg_accum_kernel<<<cdiv(E, B), B, 0, stream>>>(dst, isd, E);
    rsqrt_kernel<<<cdiv(N, B), B, 0, stream>>>(isd, N);

    // 2) h = relu(x @ projW + projb)
    {
        dim3 grid(cdiv(N, 16), cdiv(H, 128));
        gemm_wmma_kernel<<<grid, B, 0, stream>>>(x, projW, projb, h, N, FIN, H, 1);
    }

    // 3) GCN layers
    for (int l = 0; l < L; ++l) {
        fill_kernel<<<cdiv((long long)N * H, B), B, 0, stream>>>(agg, 0.0f, (long long)N * H);
        dim3 grid(cdiv(N, 16), cdiv(H, 128));
        gemm_wmma_kernel<<<grid, B, 0, stream>>>(h, convW + (size_t)l * H * H, nullptr, hW, N, H, H, 0);
        edge_kernel<<<cdiv((long long)E * (H / 4), B), B, 0, stream>>>(src, dst, isd, hW, agg, E, H);
        post_kernel<<<cdiv((long long)N * H, B), B, 0, stream>>>(
            h, hW, agg, isd, convb + (size_t)l * H,
            bng + (size_t)l * H, bnb + (size_t)l * H,
            bnm + (size_t)l * H, bnv + (size_t)l * H, N, H);
    }

    // 4) pooling: mean (cols 0..H-1) + max (cols H..2H-1); h >= 0 so uint-max and 0-init are valid
    fill_kernel<<<cdiv((long long)G * 2 * H, B), B, 0, stream>>>(g, 0.0f, (long long)G * 2 * H);
    fill_kernel<<<cdiv(G, B), B, 0, stream>>>(cnt, 0.0f, G);
    cnt_kernel<<<cdiv(N, B), B, 0, stream>>>(batch, cnt, N);
    pool_kernel<<<cdiv((long long)N * (H / 4), B), B, 0, stream>>>(batch, h, g, N, H);
    poolfin_kernel<<<cdiv((long long)G * H, B), B, 0, stream>>>(g, cnt, G, H);

    // 5) head MLPs
    {
        dim3 grid1(cdiv(G, 16), cdiv(H, 128));
        gemm_wmma_kernel<<<grid1, B, 0, stream>>>(g, W1, b1, g1, G, 2 * H, H, 1);
        dim3 grid2(cdiv(G, 16), cdiv(H2, 128));
        gemm_wmma_kernel<<<grid2, B, 0, stream>>>(g1, W2, b2, g2, G, H, H2, 1);
        head3_kernel<<<cdiv((long long)G * T, B), B, 0, stream>>>(g2, W3, b3, (float*)d_out, G, H2, T);
    }
}